// MMD_4234837754211
// MI455X (gfx1250) — compile-verified
//
#include <hip/hip_runtime.h>

typedef float v2f __attribute__((ext_vector_type(2)));
typedef float v4f __attribute__((ext_vector_type(4)));
typedef float v8f __attribute__((ext_vector_type(8)));

#define P_DIM 64          // feature dim (fixed by reference)
#define TILE 128          // distance-matrix tile per workgroup
#define LDS_STRIDE 68     // 64 + 4 floats pad -> conflict-free ds_load_b64

// ---------------------------------------------------------------------------
// Row squared-norms for X and Y in one pass.
// ---------------------------------------------------------------------------
__global__ __launch_bounds__(256) void mmd_norms_kernel(
    const float* __restrict__ X, const float* __restrict__ Y,
    int na, int nb, float* __restrict__ norms)
{
    int idx = blockIdx.x * blockDim.x + threadIdx.x;
    if (idx >= na + nb) return;
    const float* row = (idx < na) ? (X + (size_t)idx * P_DIM)
                                  : (Y + (size_t)(idx - na) * P_DIM);
    float s = 0.0f;
#pragma unroll
    for (int k = 0; k < P_DIM; k += 4) {
        v4f v = *(const v4f*)(row + k);
        s += v.x * v.x + v.y * v.y + v.z * v.z + v.w * v.w;
    }
    norms[idx] = s;
}

// ---------------------------------------------------------------------------
// Gaussian-kernel tile sum:  partial[block] = sum over assigned 128x128 tiles
// of exp(-(an[i]+bn[j]-2*A_i.B_j)/2), optionally excluding the i==j diagonal.
// GEMM on v_wmma_f32_16x16x4_f32. K is the outer loop with 8 live v8f
// accumulators so each A fragment is loaded from global exactly once per tile
// and feeds 8 back-to-back WMMAs.
// ---------------------------------------------------------------------------
__global__ __launch_bounds__(256) void mmd_gauss_tile_kernel(
    const float* __restrict__ A, const float* __restrict__ B,
    const float* __restrict__ an, const float* __restrict__ bn,
    int na, int nb, int diag, float* __restrict__ partial)
{
    __shared__ float smem[TILE * LDS_STRIDE];
    __shared__ float wred[8];

    const int tid  = threadIdx.x;
    const int wave = tid >> 5;
    const int lane = tid & 31;
    const int l15  = lane & 15;
    const int lhi  = lane >> 4;      // 0 for lanes 0-15, 1 for lanes 16-31
    const int klo  = lhi * 2;        // K offset per ISA 16x4 f32 A/B layout

    const int tiles_i = na / TILE;
    const int tiles_j = nb / TILE;
    const int ntiles  = tiles_i * tiles_j;

    float acc = 0.0f;

    for (int tt = blockIdx.x; tt < ntiles; tt += gridDim.x) {
        const int ti = tt / tiles_j;
        const int tj = tt - ti * tiles_j;
        const int i0 = ti * TILE;
        const int j0 = tj * TILE;

        // ---- stage B tile (128 rows x 64 floats) into padded LDS ----
        {
            const int r = tid >> 1;           // 0..127
            const int h = (tid & 1) * 32;     // 0 or 32
            const float* src = B + (size_t)(j0 + r) * P_DIM + h;
            float* dst = smem + r * LDS_STRIDE + h;
#pragma unroll
            for (int q = 0; q < 32; q += 4)
                *(v4f*)(dst + q) = *(const v4f*)(src + q);
        }
        __syncthreads();

        const int i0w = i0 + wave * 16;       // this wave's 16-row strip

        // ---- K-outer accumulation over 8 column sub-tiles ----
        v8f c[8];
#pragma unroll
        for (int t = 0; t < 8; ++t)
            c[t] = (v8f){0.f, 0.f, 0.f, 0.f, 0.f, 0.f, 0.f, 0.f};

        const float* ap    = A + (size_t)(i0w + l15) * P_DIM + klo;
        const float* bbase = smem + l15 * LDS_STRIDE + klo;
#pragma unroll
        for (int kk = 0; kk < 16; ++kk) {
            const v2f afrag = *(const v2f*)(ap + 4 * kk);
#pragma unroll
            for (int t = 0; t < 8; ++t) {
                const v2f bfrag =
                    *(const v2f*)(bbase + t * 16 * LDS_STRIDE + 4 * kk);
                c[t] = __builtin_amdgcn_wmma_f32_16x16x4_f32(
                           /*neg_a=*/false, afrag,
                           /*neg_b=*/false, bfrag,
                           /*c_mod=*/(short)0, c[t],
                           /*reuse_a=*/false, /*reuse_b=*/false);
            }
        }

        // ---- row norms this lane needs (C layout: M = v + 8*lhi) ----
        float xn[8];
        {
            const float* np = an + i0w + 8 * lhi;
#pragma unroll
            for (int v = 0; v < 8; ++v) xn[v] = np[v];
        }
        const int growbase = i0w + 8 * lhi;

        // ---- epilogue: d2 -> exp -> masked accumulate ----
#pragma unroll
        for (int t = 0; t < 8; ++t) {
            const int gcol = j0 + t * 16 + l15;
            const float yn = bn[gcol];
#pragma unroll
            for (int v = 0; v < 8; ++v) {
                float d2 = xn[v] + yn - 2.0f * c[t][v];
                float e  = __expf(-0.5f * d2);
                if (!(diag && (growbase + v) == gcol)) acc += e;
            }
        }
        __syncthreads();
    }

    // ---- deterministic reduction: shuffle tree -> LDS -> block sum ----
#pragma unroll
    for (int off = 16; off >= 1; off >>= 1)
        acc += __shfl_down(acc, off, 32);
    if (lane == 0) wred[wave] = acc;
    __syncthreads();
    if (tid == 0) {
        float s = 0.0f;
#pragma unroll
        for (int w = 0; w < 8; ++w) s += wred[w];
        partial[blockIdx.x] = s;
    }
}

// ---------------------------------------------------------------------------
// Final combine: fixed-order reduction of the three partial regions, then
// out = kxx/(n(n-1)) + kyy/(m(m-1)) - 2*kxy/(n*m)
// ---------------------------------------------------------------------------
__global__ __launch_bounds__(256) void mmd_combine_kernel(
    const float* __restrict__ partial, int P, int na, int nb,
    float* __restrict__ out)
{
    __shared__ float red[256];
    const int tid = threadIdx.x;
    float s[3];
    for (int r = 0; r < 3; ++r) {
        float a = 0.0f;
        for (int i = tid; i < P; i += 256) a += partial[r * P + i];
        red[tid] = a;
        __syncthreads();
        for (int st = 128; st > 0; st >>= 1) {
            if (tid < st) red[tid] += red[tid + st];
            __syncthreads();
        }
        s[r] = red[0];
        __syncthreads();
    }
    if (tid == 0) {
        double n = (double)na, m = (double)nb;
        float kxx = s[0] / (float)(n * (n - 1.0));
        float kyy = s[1] / (float)(m * (m - 1.0));
        float kxy = s[2] / (float)(n * m);
        out[0] = kxx + kyy - 2.0f * kxy;
    }
}

// ---------------------------------------------------------------------------
extern "C" void kernel_launch(void* const* d_in, const int* in_sizes, int n_in,
                              void* d_out, int out_size, void* d_ws, size_t ws_size,
                              hipStream_t stream)
{
    (void)n_in; (void)out_size;
    const float* X = (const float*)d_in[0];
    const float* Y = (const float*)d_in[1];
    const int na = in_sizes[0] / P_DIM;   // 8192
    const int nb = in_sizes[1] / P_DIM;   // 8192
    float* out = (float*)d_out;

    // workspace layout: [na+nb norms][3*P partials]
    int P = 1024;
    while (P > 1 &&
           ((size_t)(na + nb) + 3 * (size_t)P) * sizeof(float) > ws_size)
        P >>= 1;

    float* norms   = (float*)d_ws;
    float* xn      = norms;
    float* yn      = norms + na;
    float* partial = norms + (na + nb);

    const int tot = na + nb;
    mmd_norms_kernel<<<(tot + 255) / 256, 256, 0, stream>>>(X, Y, na, nb, norms);

    mmd_gauss_tile_kernel<<<P, 256, 0, stream>>>(X, X, xn, xn, na, na, 1, partial + 0 * P);
    mmd_gauss_tile_kernel<<<P, 256, 0, stream>>>(Y, Y, yn, yn, nb, nb, 1, partial + 1 * P);
    mmd_gauss_tile_kernel<<<P, 256, 0, stream>>>(X, Y, xn, yn, na, nb, 0, partial + 2 * P);

    mmd_combine_kernel<<<1, 256, 0, stream>>>(partial, P, na, nb, out);
}